// GATConv_19731079758622
// MI455X (gfx1250) — compile-verified
//
#include <hip/hip_runtime.h>

typedef float v2f __attribute__((ext_vector_type(2)));
typedef float v8f __attribute__((ext_vector_type(8)));

#define DIM_IN   128
#define DIM_OUT  64
#define NEG_SLOPE 0.2f

// ---- order-preserving float<->uint encoding for atomicMax on floats ----
__device__ __forceinline__ unsigned enc_f32(float f) {
    unsigned u = __float_as_uint(f);
    return (u & 0x80000000u) ? ~u : (u | 0x80000000u);
}
__device__ __forceinline__ float dec_f32(unsigned e) {
    unsigned u = (e & 0x80000000u) ? (e ^ 0x80000000u) : ~e;
    return __uint_as_float(u);
}

// ------------------------------------------------------------------
// Kernel 1: h = x @ W via V_WMMA_F32_16X16X4_F32 (fp32, K=4 per issue)
// Each wave computes a 16x64 tile; block of 256 threads = 8 waves = 128 rows.
// Tail rows are handled by CLAMPING the A-row index (garbage only feeds
// output rows we never store), keeping all loads unconditional so the
// K-loop has no exec-mask branches and A loads merge into global_load_b64.
// ------------------------------------------------------------------
__global__ __launch_bounds__(256) void gat_gemm_wmma(
    const float* __restrict__ x, const float* __restrict__ W,
    float* __restrict__ h, int N)
{
    const int lane = threadIdx.x & 31;
    const int wave = threadIdx.x >> 5;
    const int half = lane >> 4;     // 0: lanes 0-15, 1: lanes 16-31
    const int lm   = lane & 15;
    const int m0   = (blockIdx.x * 8 + wave) * 16;
    if (m0 >= N) return;            // wave-uniform guard: EXEC stays full

    int row = m0 + lm;
    if (row > N - 1) row = N - 1;   // clamp: unconditional loads, no divergence
    const float* xrow = x + (size_t)row * DIM_IN;

    v8f acc0 = {}, acc1 = {}, acc2 = {}, acc3 = {};

    #pragma unroll 4
    for (int k = 0; k < DIM_IN; k += 4) {
        const int ka = k + 2 * half;                 // K index this half-wave owns
        const v2f a = *(const v2f*)(xrow + ka);      // A VGPR0: K=ka, VGPR1: K=ka+1

        const float* wr0 = W + (size_t)ka * DIM_OUT; // row K=ka
        const float* wr1 = wr0 + DIM_OUT;            // row K=ka+1
        v2f b0, b1, b2, b3;
        b0.x = wr0[ 0 + lm]; b0.y = wr1[ 0 + lm];
        b1.x = wr0[16 + lm]; b1.y = wr1[16 + lm];
        b2.x = wr0[32 + lm]; b2.y = wr1[32 + lm];
        b3.x = wr0[48 + lm]; b3.y = wr1[48 + lm];

        acc0 = __builtin_amdgcn_wmma_f32_16x16x4_f32(false, a, false, b0, (short)0, acc0, false, false);
        acc1 = __builtin_amdgcn_wmma_f32_16x16x4_f32(false, a, false, b1, (short)0, acc1, false, false);
        acc2 = __builtin_amdgcn_wmma_f32_16x16x4_f32(false, a, false, b2, (short)0, acc2, false, false);
        acc3 = __builtin_amdgcn_wmma_f32_16x16x4_f32(false, a, false, b3, (short)0, acc3, false, false);
    }

    // D layout: VGPR i -> M = m0 + i + 8*half, N = colTile*16 + lm
    #pragma unroll
    for (int i = 0; i < 8; ++i) {
        const int r = m0 + i + 8 * half;
        if (r < N) {
            float* hr = h + (size_t)r * DIM_OUT;
            hr[ 0 + lm] = acc0[i];
            hr[16 + lm] = acc1[i];
            hr[32 + lm] = acc2[i];
            hr[48 + lm] = acc3[i];
        }
    }
}

// ------------------------------------------------------------------
// Kernel 2: a_src[n] = h[n,:]·att_src ; a_dst[n] = h[n,:]·att_dst
// One wave per node; 2 elems per lane + xor-shuffle reduction.
// ------------------------------------------------------------------
__global__ __launch_bounds__(256) void gat_node_att(
    const float* __restrict__ h, const float* __restrict__ att_src,
    const float* __restrict__ att_dst, float* __restrict__ a_src,
    float* __restrict__ a_dst, int N)
{
    const int lane = threadIdx.x & 31;
    const int wid  = (blockIdx.x * blockDim.x + threadIdx.x) >> 5;
    const int nw   = (gridDim.x * blockDim.x) >> 5;
    const float as0 = att_src[lane], as1 = att_src[lane + 32];
    const float ad0 = att_dst[lane], ad1 = att_dst[lane + 32];
    for (int n = wid; n < N; n += nw) {
        const float v0 = h[(size_t)n * DIM_OUT + lane];
        const float v1 = h[(size_t)n * DIM_OUT + lane + 32];
        float s = v0 * as0 + v1 * as1;
        float d = v0 * ad0 + v1 * ad1;
        #pragma unroll
        for (int off = 16; off > 0; off >>= 1) {
            s += __shfl_xor(s, off, 32);
            d += __shfl_xor(d, off, 32);
        }
        if (lane == 0) { a_src[n] = s; a_dst[n] = d; }
    }
}

// ------------------------------------------------------------------
// Kernel 3: zero out / denom / emax (d_out + ws are poisoned by harness)
// ------------------------------------------------------------------
__global__ void gat_init(float* __restrict__ out, float* __restrict__ denom,
                         unsigned* __restrict__ emax, int N)
{
    const size_t i = (size_t)blockIdx.x * blockDim.x + threadIdx.x;
    if (i < (size_t)N * DIM_OUT) out[i] = 0.0f;
    if (i < (size_t)N) { denom[i] = 0.0f; emax[i] = 0u; }  // 0 < enc of any finite
}

// ------------------------------------------------------------------
// Kernel 4: e = leaky_relu(a_src[src]+a_dst[dst]); segment max via atomicMax
// Edges i>=E are the self loops (src=dst=i-E).
// ------------------------------------------------------------------
__global__ void gat_edge_max(const int* __restrict__ srcs, const int* __restrict__ dsts,
                             const float* __restrict__ a_src, const float* __restrict__ a_dst,
                             float* __restrict__ e_buf, unsigned* __restrict__ emax,
                             int E, int ET)
{
    const int i = blockIdx.x * blockDim.x + threadIdx.x;
    if (i >= ET) return;
    const int s = (i < E) ? srcs[i] : (i - E);
    const int d = (i < E) ? dsts[i] : (i - E);
    float e = a_src[s] + a_dst[d];
    e = (e > 0.0f) ? e : NEG_SLOPE * e;
    e_buf[i] = e;
    atomicMax(&emax[d], enc_f32(e));
}

// ------------------------------------------------------------------
// Kernel 5: e_exp = exp(e - emax[dst]); denom[dst] += e_exp (in-place over e)
// ------------------------------------------------------------------
__global__ void gat_edge_exp(const int* __restrict__ dsts, float* __restrict__ e_buf,
                             const unsigned* __restrict__ emax, float* __restrict__ denom,
                             int E, int ET)
{
    const int i = blockIdx.x * blockDim.x + threadIdx.x;
    if (i >= ET) return;
    const int d = (i < E) ? dsts[i] : (i - E);
    const float ex = __expf(e_buf[i] - dec_f32(emax[d]));
    e_buf[i] = ex;
    atomicAdd(&denom[d], ex);
}

// ------------------------------------------------------------------
// Kernel 6: out[dst,:] += (e_exp/denom[dst]) * h[src,:]  (wave per edge,
// 2 f32 per lane; h/out are L2-resident so atomics resolve in L2)
// ------------------------------------------------------------------
__global__ __launch_bounds__(256) void gat_aggregate(
    const int* __restrict__ srcs, const int* __restrict__ dsts,
    const float* __restrict__ e_exp, const float* __restrict__ denom,
    const float* __restrict__ h, float* __restrict__ out, int E, int ET)
{
    const int lane = threadIdx.x & 31;
    const int wid  = (blockIdx.x * blockDim.x + threadIdx.x) >> 5;
    const int nw   = (gridDim.x * blockDim.x) >> 5;
    for (int i = wid; i < ET; i += nw) {
        const int s = (i < E) ? srcs[i] : (i - E);
        const int d = (i < E) ? dsts[i] : (i - E);
        const float alpha = e_exp[i] / (denom[d] + 1e-16f);
        const float v0 = h[(size_t)s * DIM_OUT + lane]      * alpha;
        const float v1 = h[(size_t)s * DIM_OUT + lane + 32] * alpha;
        atomicAdd(&out[(size_t)d * DIM_OUT + lane],      v0);
        atomicAdd(&out[(size_t)d * DIM_OUT + lane + 32], v1);
    }
}

// ------------------------------------------------------------------
extern "C" void kernel_launch(void* const* d_in, const int* in_sizes, int n_in,
                              void* d_out, int out_size, void* d_ws, size_t ws_size,
                              hipStream_t stream)
{
    const float* x        = (const float*)d_in[0];
    const float* W        = (const float*)d_in[1];
    const float* att_src  = (const float*)d_in[2];
    const float* att_dst  = (const float*)d_in[3];
    const int*   edge_idx = (const int*)d_in[4];

    const int N  = in_sizes[0] / DIM_IN;
    const int E  = in_sizes[4] / 2;
    const int ET = E + N;
    const int* srcs = edge_idx;       // edge_index[0], row-major (2,E)
    const int* dsts = edge_idx + E;   // edge_index[1]
    float* out = (float*)d_out;

    // workspace layout (floats): h | a_src | a_dst | e | denom | emax
    float*    h      = (float*)d_ws;
    float*    a_src  = h + (size_t)N * DIM_OUT;
    float*    a_dst  = a_src + N;
    float*    e_buf  = a_dst + N;
    float*    denom  = e_buf + ET;
    unsigned* emax   = (unsigned*)(denom + N);

    const int outElems = N * DIM_OUT;
    gat_init     <<<(outElems + 255) / 256, 256, 0, stream>>>(out, denom, emax, N);
    gat_gemm_wmma<<<(N + 127) / 128,        256, 0, stream>>>(x, W, h, N);
    gat_node_att <<<1024,                   256, 0, stream>>>(h, att_src, att_dst, a_src, a_dst, N);
    gat_edge_max <<<(ET + 255) / 256,       256, 0, stream>>>(srcs, dsts, a_src, a_dst, e_buf, emax, E, ET);
    gat_edge_exp <<<(ET + 255) / 256,       256, 0, stream>>>(dsts, e_buf, emax, denom, E, ET);
    gat_aggregate<<<2048,                   256, 0, stream>>>(srcs, dsts, e_buf, denom, h, out, E, ET);
}